// Convolution_16441134809109
// MI455X (gfx1250) — compile-verified
//
#include <hip/hip_runtime.h>
#include <hip/hip_bf16.h>
#include <stdint.h>

#define G 128
#define NCELL (63 * 63 * 63)              // 250047 possible downsampled cells
#define NCELL_AL 250048
#define SCAN_NB ((NCELL + 255) / 256)     // 977 scan blocks
#define NTILES ((NCELL + 15) / 16)        // 15628 max 16-row WMMA tiles

typedef __attribute__((ext_vector_type(2))) float v2f;
typedef __attribute__((ext_vector_type(8))) float v8f;

// ---- init: out rows = 0, out_pos rows = -0.5 (pad defaults) ----------------
__global__ void init_out_kernel(float* __restrict__ out, int n_out16, int n_total) {
  int idx = blockIdx.x * blockDim.x + threadIdx.x;
  if (idx >= n_total) return;
  out[idx] = (idx < n_out16) ? 0.0f : -0.5f;
}

// ---- scatter features into dense voxel grid + mark occupancy ---------------
__global__ void scatter_kernel(const float* __restrict__ feat,
                               const int* __restrict__ pos,
                               float* __restrict__ vox,
                               unsigned* __restrict__ occ, int N) {
  int tid = blockIdx.x * blockDim.x + threadIdx.x;
  if (tid >= N * 16) return;
  int i = tid >> 4, c = tid & 15;
  int x = pos[i * 3 + 0], y = pos[i * 3 + 1], z = pos[i * 3 + 2];
  size_t vid = ((size_t)x * G + y) * G + z;
  atomicAdd(vox + vid * 16 + c, feat[tid]);
  if (c == 0) {
    unsigned cid = (unsigned)(((x >> 1) * 63 + (y >> 1)) * 63 + (z >> 1));
    occ[cid] = 1u;
  }
}

// ---- two-level exclusive scan over occupancy (sorted-unique ranks) ---------
__global__ void scan1_kernel(const unsigned* __restrict__ occ,
                             unsigned* __restrict__ rank,
                             unsigned* __restrict__ bsum) {
  __shared__ unsigned s[256];
  unsigned t = threadIdx.x;
  unsigned gid = blockIdx.x * 256u + t;
  unsigned v = (gid < NCELL) ? occ[gid] : 0u;
  s[t] = v;
  __syncthreads();
  for (unsigned off = 1; off < 256; off <<= 1) {
    unsigned x = (t >= off) ? s[t - off] : 0u;
    __syncthreads();
    s[t] += x;
    __syncthreads();
  }
  if (gid < NCELL) rank[gid] = s[t] - v;  // block-local exclusive
  if (t == 255) bsum[blockIdx.x] = s[255];
}

__global__ void scan2_kernel(const unsigned* __restrict__ bsum,
                             unsigned* __restrict__ boff,
                             unsigned* __restrict__ dM, int nb) {
  __shared__ unsigned s[1024];
  unsigned t = threadIdx.x;
  unsigned v = (t < (unsigned)nb) ? bsum[t] : 0u;
  s[t] = v;
  __syncthreads();
  for (unsigned off = 1; off < 1024; off <<= 1) {
    unsigned x = (t >= off) ? s[t - off] : 0u;
    __syncthreads();
    s[t] += x;
    __syncthreads();
  }
  if (t < (unsigned)nb) boff[t] = s[t] - v;
  if (t == 1023) *dM = s[1023];           // total unique count M
}

// ---- compact: sorted eid list + decoded output positions -------------------
__global__ void compact_kernel(const unsigned* __restrict__ occ,
                               const unsigned* __restrict__ rank,
                               const unsigned* __restrict__ boff,
                               unsigned* __restrict__ ulist,
                               float* __restrict__ out_pos) {
  unsigned cid = blockIdx.x * 256u + threadIdx.x;
  if (cid >= NCELL) return;
  if (!occ[cid]) return;
  unsigned r = rank[cid] + boff[cid >> 8];
  unsigned cx = cid / 3969u;
  unsigned rem = cid - cx * 3969u;
  unsigned cy = rem / 63u;
  unsigned cz = rem - cy * 63u;
  ulist[r] = ((cx * 2u) * G + cy * 2u) * G + cz * 2u;
  // (e + 0.5) / 2 with e = 2*c  ->  c + 0.25
  out_pos[(size_t)r * 3 + 0] = (float)cx + 0.25f;
  out_pos[(size_t)r * 3 + 1] = (float)cy + 0.25f;
  out_pos[(size_t)r * 3 + 2] = (float)cz + 0.25f;
}

// ---- main sparse-conv GEMM: 27 neighbor gathers x f32 WMMA -----------------
// One wave owns a tile of 16 output voxels. A: 16x4 f32 (lane=row, hi half ->
// K=2,3). B: 4x16 f32 from LDS-resident W. D accumulates 16x16 f32 over
// 27 offsets x 4 K-chunks = 108 v_wmma_f32_16x16x4_f32 per tile.
__global__ void __launch_bounds__(256)
sparse_conv_wmma_kernel(const float* __restrict__ vox,
                        const float* __restrict__ Wg,
                        const unsigned* __restrict__ ulist,
                        const unsigned* __restrict__ dM,
                        float* __restrict__ out) {
  __shared__ float ldsW[27 * 256];
  for (int idx = threadIdx.x; idx < 27 * 256; idx += 256) ldsW[idx] = Wg[idx];
  __syncthreads();

  unsigned M = *dM;
  unsigned tile = blockIdx.x * 8u + (threadIdx.x >> 5);
  unsigned r0 = tile * 16u;
  if (r0 >= M) return;                    // wave-uniform exit; EXEC stays full

  unsigned lane = threadIdx.x & 31u;
  unsigned m = lane & 15u;                // A row / B,D column
  unsigned hi = lane >> 4;                // selects K pair {0,1} vs {2,3}
  unsigned r = r0 + m;
  unsigned rc = (r < M) ? r : (M - 1u);   // clamp: never touch stale ulist
  unsigned eid = ulist[rc];
  const float* rowBase = vox + (size_t)eid * 16u + 2u * hi;
  const float* wBase = ldsW + (size_t)(2u * hi) * 16u + m;

  v8f acc = {0.f, 0.f, 0.f, 0.f, 0.f, 0.f, 0.f, 0.f};

  int k = 0;
  for (int dx = 0; dx < 3; ++dx)
    for (int dy = 0; dy < 3; ++dy)
      for (int dz = 0; dz < 3; ++dz, ++k) {
        const float* ap = rowBase + (size_t)((dx * G + dy) * G + dz) * 16u;
        const float* wp = wBase + k * 256;
#pragma unroll
        for (int c = 0; c < 4; ++c) {
          v2f a = *(const v2f*)(ap + 4 * c);      // K=4c+2hi, 4c+2hi+1
          v2f b;
          b.x = wp[c * 64];                        // W[k][4c+2hi][n]
          b.y = wp[c * 64 + 16];                   // W[k][4c+2hi+1][n]
          acc = __builtin_amdgcn_wmma_f32_16x16x4_f32(
              false, a, false, b, (short)0, acc, false, false);
        }
      }

#pragma unroll
  for (int j = 0; j < 8; ++j) {
    unsigned row = r0 + (unsigned)j + 8u * hi;     // D: VGPR j -> rows j, j+8
    if (row < M) out[(size_t)row * 16u + m] = acc[j];
  }
}

extern "C" void kernel_launch(void* const* d_in, const int* in_sizes, int n_in,
                              void* d_out, int out_size, void* d_ws, size_t ws_size,
                              hipStream_t stream) {
  const float* feat = (const float*)d_in[0];   // [N,16] f32
  const int* pos = (const int*)d_in[1];        // [N,3] i32
  const float* Wg = (const float*)d_in[2];     // [27,16,16] f32
  float* out = (float*)d_out;                  // [N*16] out ++ [N*3] out_pos
  const int N = in_sizes[0] / 16;

  // Workspace layout
  float* vox = (float*)d_ws;                              // 128^3 * 16 f32 = 128 MB
  size_t voxElems = (size_t)G * G * G * 16;
  unsigned* occ = (unsigned*)(vox + voxElems);            // NCELL
  unsigned* rank = occ + NCELL_AL;                        // NCELL
  unsigned* bsum = rank + NCELL_AL;                       // SCAN_NB (pad 1024)
  unsigned* boff = bsum + 1024;                           // SCAN_NB (pad 1024)
  unsigned* dM = boff + 1024;                             // 1
  unsigned* ulist = dM + 16;                              // NCELL

  // Zero grid + occupancy (accumulated into every call)
  hipMemsetAsync(d_ws, 0, voxElems * sizeof(float) + NCELL_AL * sizeof(unsigned),
                 stream);

  int n_total = N * 19;  // 16 out channels + 3 pos per row
  init_out_kernel<<<(n_total + 255) / 256, 256, 0, stream>>>(out, N * 16, n_total);

  scatter_kernel<<<(N * 16 + 255) / 256, 256, 0, stream>>>(feat, pos, vox, occ, N);

  scan1_kernel<<<SCAN_NB, 256, 0, stream>>>(occ, rank, bsum);
  scan2_kernel<<<1, 1024, 0, stream>>>(bsum, boff, dM, SCAN_NB);
  compact_kernel<<<SCAN_NB, 256, 0, stream>>>(occ, rank, boff, ulist,
                                              out + (size_t)N * 16);

  sparse_conv_wmma_kernel<<<(NTILES + 7) / 8, 256, 0, stream>>>(vox, Wg, ulist,
                                                                dM, out);
}